// RESAGEConv_74852690035476
// MI455X (gfx1250) — compile-verified
//
#include <hip/hip_runtime.h>
#include <hip/hip_bf16.h>

typedef __attribute__((ext_vector_type(2))) float v2f;
typedef __attribute__((ext_vector_type(8))) float v8f;

#define IN_FEATS  128
#define OUT_FEATS 64
#define ALPHA     10.0f

__device__ __forceinline__ float edge_w(const float* __restrict__ edge_weight, int etype) {
    // leaky_relu(edge_weight * ALPHA), slope 0.01; e_feat is 1-based
    float x = edge_weight[etype - 1] * ALPHA;
    return x > 0.0f ? x : 0.01f * x;
}

// Guaranteed hardware f32 atomic add (no-return form, tracked by STOREcnt;
// implicit s_wait_idle at s_endpgm drains it). Avoids any CAS-loop expansion.
__device__ __forceinline__ void atomic_add_f32(float* p, float v) {
    asm volatile("global_atomic_add_f32 %0, %1, off"
                 :
                 : "v"(p), "v"(v)
                 : "memory");
}

// ---------------- Kernel 1: zero degree accumulator ----------------
__global__ void resage_zero_deg(float* __restrict__ deg, int n) {
    int i = blockIdx.x * blockDim.x + threadIdx.x;
    if (i < n) deg[i] = 0.0f;
}

// ---------------- Kernel 2: deg[dst] += ew(e) ----------------------
__global__ void resage_deg_accum(const int* __restrict__ dst,
                                 const int* __restrict__ e_feat,
                                 const float* __restrict__ edge_weight,
                                 float* __restrict__ deg, int E) {
    int e = blockIdx.x * blockDim.x + threadIdx.x;
    if (e >= E) return;
    atomic_add_f32(&deg[dst[e]], edge_w(edge_weight, e_feat[e]));
}

// ---------------- Kernel 3: WMMA GEMM + epilogue -------------------
// out[g,:] = (feat @ W)[g,:] + bias ;  h[g,:] = (feat @ W)[g,:] * norm[g]
// Block: 256 threads = 8 waves; covers 32 rows x 64 cols (2x4 tiles of 16x16).
__global__ void __launch_bounds__(256)
resage_gemm(const float* __restrict__ feat,
            const float* __restrict__ weight,   // [128,64] row-major
            const float* __restrict__ bias,     // [64]
            const float* __restrict__ deg,      // [N]
            float* __restrict__ out,            // [N,64]
            float* __restrict__ h,              // [N,64]
            int N) {
    __shared__ float sA[32 * IN_FEATS];          // 16 KB feat tile
    __shared__ float sW[IN_FEATS * OUT_FEATS];   // 32 KB weight
    __shared__ float snorm[32];

    const int tid  = threadIdx.x;
    const int r0   = blockIdx.x * 32;

    // stage full weight matrix (8192 floats = 2048 float4)
    {
        const float4* w4 = (const float4*)weight;
        float4* s4 = (float4*)sW;
        #pragma unroll
        for (int i = 0; i < 8; ++i) s4[tid + i * 256] = w4[tid + i * 256];
    }
    // stage 32x128 feat tile (1024 float4), zero-pad past N
    {
        const float4* f4 = (const float4*)feat;
        float4* s4 = (float4*)sA;
        #pragma unroll
        for (int i = 0; i < 4; ++i) {
            int idx4 = tid + i * 256;          // float4 index within tile
            int row  = idx4 >> 5;              // /32 float4 per row
            int g    = r0 + row;
            float4 v = make_float4(0.f, 0.f, 0.f, 0.f);
            if (g < N) v = f4[(size_t)g * 32 + (idx4 & 31)];
            s4[idx4] = v;
        }
    }
    // per-row norm = 1/max(deg,1)
    if (tid < 32) {
        int g = r0 + tid;
        float d = (g < N) ? deg[g] : 1.0f;
        snorm[tid] = 1.0f / fmaxf(d, 1.0f);
    }
    __syncthreads();

    const int wave = tid >> 5;
    const int lane = tid & 31;
    const int row_tile = wave >> 2;        // 0..1
    const int col_tile = wave & 3;         // 0..3
    const int half  = lane >> 4;           // 0 or 1
    const int l16   = lane & 15;
    const int koff  = half * 2;            // K pair select for A and B
    const int cb    = col_tile * 16;

    const float* Arow = &sA[(row_tile * 16 + l16) * IN_FEATS + koff];

    v8f c = {};
    #pragma unroll 8
    for (int k0 = 0; k0 < IN_FEATS; k0 += 4) {
        v2f a, b;
        a.x = Arow[k0];
        a.y = Arow[k0 + 1];
        b.x = sW[(k0 + koff)     * OUT_FEATS + cb + l16];
        b.y = sW[(k0 + koff + 1) * OUT_FEATS + cb + l16];
        // (neg_a, A, neg_b, B, c_mod, C, reuse_a, reuse_b)
        c = __builtin_amdgcn_wmma_f32_16x16x4_f32(false, a, false, b,
                                                  (short)0, c, false, false);
    }

    const int col = cb + l16;
    const float bv = bias[col];
    #pragma unroll
    for (int vi = 0; vi < 8; ++vi) {
        int rlocal = row_tile * 16 + vi + half * 8;
        int g = r0 + rlocal;
        if (g < N) {
            float fr = c[vi];
            size_t o = (size_t)g * OUT_FEATS + col;
            out[o] = fr + bv;
            h[o]   = fr * snorm[rlocal];
        }
    }
}

// ---------------- Kernel 4: edge scatter ---------------------------
// 16 threads per edge, float4 of h[src], 4 atomic adds into out[dst].
// h (25.6 MB) and out (25.6 MB) are L2-resident on MI455X (192 MB L2),
// so gathers and atomics resolve in L2.
__global__ void resage_scatter(const int* __restrict__ src,
                               const int* __restrict__ dst,
                               const int* __restrict__ e_feat,
                               const float* __restrict__ edge_weight,
                               const float* __restrict__ h,   // [N,64]
                               float* __restrict__ out,       // [N,64]
                               int E) {
    long long idx = (long long)blockIdx.x * blockDim.x + threadIdx.x;
    long long total = (long long)E * 16;
    if (idx >= total) return;
    int e  = (int)(idx >> 4);
    int q  = (int)(idx & 15);       // which float4 of the 64 feats
    int s  = src[e];
    int d  = dst[e];
    float w = edge_w(edge_weight, e_feat[e]);

    const float4* h4 = (const float4*)h;
    float4 hv = h4[(size_t)s * 16 + q];
    float* o = &out[(size_t)d * OUT_FEATS + q * 4];
    atomic_add_f32(o + 0, w * hv.x);
    atomic_add_f32(o + 1, w * hv.y);
    atomic_add_f32(o + 2, w * hv.z);
    atomic_add_f32(o + 3, w * hv.w);
}

extern "C" void kernel_launch(void* const* d_in, const int* in_sizes, int n_in,
                              void* d_out, int out_size, void* d_ws, size_t ws_size,
                              hipStream_t stream) {
    const float* feat        = (const float*)d_in[0];
    const float* edge_weight = (const float*)d_in[1];
    const float* weight      = (const float*)d_in[2];
    const float* bias        = (const float*)d_in[3];
    const int*   src         = (const int*)d_in[4];
    const int*   dst         = (const int*)d_in[5];
    const int*   e_feat      = (const int*)d_in[6];
    float* out = (float*)d_out;

    const int N = in_sizes[0] / IN_FEATS;
    const int E = in_sizes[4];

    // workspace: deg [N] floats, then h [N,64] floats (256B aligned)
    float* deg = (float*)d_ws;
    size_t h_off = ((size_t)N * sizeof(float) + 255) & ~(size_t)255;
    float* h = (float*)((char*)d_ws + h_off);

    resage_zero_deg<<<(N + 255) / 256, 256, 0, stream>>>(deg, N);
    resage_deg_accum<<<(E + 255) / 256, 256, 0, stream>>>(dst, e_feat, edge_weight, deg, E);
    resage_gemm<<<(N + 31) / 32, 256, 0, stream>>>(feat, weight, bias, deg, out, h, N);
    long long total = (long long)E * 16;
    resage_scatter<<<(unsigned)((total + 255) / 256), 256, 0, stream>>>(
        src, dst, e_feat, edge_weight, h, out, E);
}